// StateTrajectoryNetworkFF_9165460209988
// MI455X (gfx1250) — compile-verified
//
#include <hip/hip_runtime.h>
#include <math.h>
#include <stdint.h>

// ---------------------------------------------------------------------------
// StateTrajectoryNetwork scan:  out[t] = state[t+1] = state[t] + MLP(x[t], state[t])
// T = 2048 sequential steps, B = 16384 independent lanes.
//   - one thread per batch element, f32 throughout (matches reference dtype)
//   - native gfx1250 v_tanh_f32 for the 12 tanh/step (trans-bound kernel)
//   - 8-deep async global->LDS prefetch pipeline for x (ASYNCcnt), per-lane slots
//   - main loop prefetches unconditionally; 8-step epilogue drains (no clamp SALU)
//   - nontemporal streaming stores for out
// ---------------------------------------------------------------------------

#define T_LEN   2048
#define B_LEN   16384
#define BLOCK   32          // single-wave workgroups: 512 of them, max spread
#define STAGES  8           // async prefetch depth (1 KB LDS per block)

#if defined(__gfx1250__) && __has_builtin(__builtin_amdgcn_tanhf)
#define FAST_TANH(v) __builtin_amdgcn_tanhf(v)   // v_tanh_f32 (gfx1250 trans op)
#else
#define FAST_TANH(v) tanhf(v)
#endif

// Async copy of 4 bytes from global memory at base plus gbl_byte_off to the LDS address in lds_byte_off.
// Tracked with ASYNCcnt; async loads complete in order.
__device__ __forceinline__ void async_load_x(const float* __restrict__ base,
                                             uint32_t gbl_byte_off,
                                             uint32_t lds_byte_off) {
    asm volatile("global_load_async_to_lds_b32 %0, %1, %2"
                 :
                 : "v"(lds_byte_off), "v"(gbl_byte_off), "s"(base)
                 : "memory");
}

__device__ __forceinline__ void wait_async(int n) {
    if (n == 0)
        asm volatile("s_wait_asynccnt 0" ::: "memory");
    else
        asm volatile("s_wait_asynccnt %0" :: "i"(STAGES - 1) : "memory");
}

__device__ __forceinline__ void wait_ds0(void) {
    asm volatile("s_wait_dscnt 0" ::: "memory");
}

struct Weights {
    float w1[4][2], w2[4][4], w3[4][4], w4[4], bias2[4];
};

// One timestep of the MLP; tree-shaped dots to shorten the serial chain.
__device__ __forceinline__ float mlp_step(const Weights& W, float xv, float s) {
    float h1[4], h2[4], h3[4];
#pragma unroll
    for (int i = 0; i < 4; ++i)
        h1[i] = FAST_TANH(fmaf(W.w1[i][1], s, W.w1[i][0] * xv));
#pragma unroll
    for (int i = 0; i < 4; ++i) {
        float p = fmaf(W.w2[i][0], h1[0], W.w2[i][1] * h1[1]);
        float q = fmaf(W.w2[i][2], h1[2], fmaf(W.w2[i][3], h1[3], W.bias2[i]));
        h2[i] = FAST_TANH(p + q);
    }
#pragma unroll
    for (int i = 0; i < 4; ++i) {
        float p = fmaf(W.w3[i][0], h2[0], W.w3[i][1] * h2[1]);
        float q = fmaf(W.w3[i][2], h2[2], W.w3[i][3] * h2[3]);
        h3[i] = FAST_TANH(p + q);
    }
    float p = fmaf(W.w4[0], h3[0], W.w4[1] * h3[1]);
    float q = fmaf(W.w4[2], h3[2], fmaf(W.w4[3], h3[3], s));   // + residual state
    return p + q;
}

__launch_bounds__(BLOCK, 1)
__global__ void stn_scan_kernel(const float* __restrict__ x,
                                const float* __restrict__ W1,   // (4,2) row-major
                                const float* __restrict__ W2,   // (4,4)
                                const float* __restrict__ b2,   // (4)
                                const float* __restrict__ W3,   // (4,4)
                                const float* __restrict__ W4,   // (1,4)
                                float* __restrict__ out)
{
    __shared__ float xbuf[STAGES][BLOCK];

    const int tid = threadIdx.x;
    const int b   = blockIdx.x * BLOCK + tid;

    // ---- weights: uniform -> compiler promotes to scalar loads / SGPRs
    Weights W;
#pragma unroll
    for (int i = 0; i < 4; ++i) {
        W.w1[i][0] = W1[i * 2 + 0];
        W.w1[i][1] = W1[i * 2 + 1];
        W.bias2[i] = b2[i];
        W.w4[i]    = W4[i];
#pragma unroll
        for (int j = 0; j < 4; ++j) {
            W.w2[i][j] = W2[i * 4 + j];
            W.w3[i][j] = W3[i * 4 + j];
        }
    }

    // per-lane LDS slot byte addresses for the rotating stages
    uint32_t lds_off[STAGES];
#pragma unroll
    for (int k = 0; k < STAGES; ++k)
        lds_off[k] = (uint32_t)(uintptr_t)&xbuf[k][tid];

    const uint32_t col_byte = (uint32_t)b * 4u;        // byte offset of this lane's column
    const uint32_t row_bytes = (uint32_t)B_LEN * 4u;   // bytes per timestep row

    // ---- prologue: fill the pipeline with x[0..STAGES-1]
#pragma unroll
    for (int k = 0; k < STAGES; ++k)
        async_load_x(x, (uint32_t)k * row_bytes + col_byte, lds_off[k]);

    float s = 0.0f;                                    // recurrent state (zeros)

    // ---- main loop: steps [0, T-STAGES) with unconditional prefetch.
    // Invariant at iter t: issued = t + STAGES, so asynccnt<=STAGES-1 => load t done.
    for (int t = 0; t < T_LEN - STAGES; ++t) {
        const int stage = t & (STAGES - 1);

        wait_async(STAGES - 1);
        float xv = xbuf[stage][tid];                   // ds_load_b32 (own slot)
        wait_ds0();                                    // value in reg before slot reuse
        async_load_x(x, (uint32_t)(t + STAGES) * row_bytes + col_byte, lds_off[stage]);

        float o = mlp_step(W, xv, s);
        __builtin_nontemporal_store(o, &out[(uint32_t)t * B_LEN + (uint32_t)b]);
        s = o;
    }

    // ---- epilogue: everything issued; drain once, no more prefetch.
    wait_async(0);
#pragma unroll
    for (int t = T_LEN - STAGES; t < T_LEN; ++t) {
        float xv = xbuf[t & (STAGES - 1)][tid];
        float o  = mlp_step(W, xv, s);
        __builtin_nontemporal_store(o, &out[(uint32_t)t * B_LEN + (uint32_t)b]);
        s = o;
    }
}

extern "C" void kernel_launch(void* const* d_in, const int* in_sizes, int n_in,
                              void* d_out, int out_size, void* d_ws, size_t ws_size,
                              hipStream_t stream) {
    (void)in_sizes; (void)n_in; (void)out_size; (void)d_ws; (void)ws_size;
    // setup_inputs() dict order: x, W1, W2, b2, W3, W4
    const float* x  = (const float*)d_in[0];
    const float* W1 = (const float*)d_in[1];
    const float* W2 = (const float*)d_in[2];
    const float* b2 = (const float*)d_in[3];
    const float* W3 = (const float*)d_in[4];
    const float* W4 = (const float*)d_in[5];
    float* out = (float*)d_out;

    dim3 grid(B_LEN / BLOCK);
    dim3 block(BLOCK);
    stn_scan_kernel<<<grid, block, 0, stream>>>(x, W1, W2, b2, W3, W4, out);
}